// CustomGINE_5970004542027
// MI455X (gfx1250) — compile-verified
//
#include <hip/hip_runtime.h>

#define N_NODES 100000
#define N_EDGES 1600000
#define DIM     128
#define LN_EPS  1e-5f

typedef __attribute__((ext_vector_type(2))) float v2f;
typedef __attribute__((ext_vector_type(8))) float v8f;

// ---------------------------------------------------------------------------
// Kernel 1: h = (1 + eps) * x   (streaming float4, also initializes scatter dst)
// ---------------------------------------------------------------------------
__global__ void k_init(const float* __restrict__ x,
                       const float* __restrict__ eps,
                       float* __restrict__ h) {
  long long i  = (long long)blockIdx.x * blockDim.x + threadIdx.x;
  long long n4 = (long long)N_NODES * DIM / 4;
  if (i >= n4) return;
  float s = 1.0f + eps[0];
  float4 v = ((const float4*)x)[i];
  v.x *= s; v.y *= s; v.z *= s; v.w *= s;
  ((float4*)h)[i] = v;
}

// ---------------------------------------------------------------------------
// Kernel 2: per-edge message + scatter-add.
// One wave32 per edge; lane L owns floats [4L, 4L+4) of the 128-dim feature.
// msg = relu(x[src] + edge_emb[attr]);  atomicAdd into h[dst].
// ---------------------------------------------------------------------------
__global__ void k_scatter(const float* __restrict__ x,
                          const int*   __restrict__ ei,   // [2, E]
                          const int*   __restrict__ ea,   // [E]
                          const float* __restrict__ emb,  // [4, 128]
                          float* __restrict__ h) {
  long long gt = (long long)blockIdx.x * blockDim.x + threadIdx.x;
  int e = (int)(gt >> 5);
  if (e >= N_EDGES) return;
  int lane = threadIdx.x & 31;

  int src = ei[e];
  int dst = ei[N_EDGES + e];
  int a   = ea[e];

  float4 xv = *(const float4*)(x   + (long long)src * DIM + lane * 4);
  float4 ev = *(const float4*)(emb + (long long)a   * DIM + lane * 4);
  float4 m;
  m.x = fmaxf(xv.x + ev.x, 0.0f);
  m.y = fmaxf(xv.y + ev.y, 0.0f);
  m.z = fmaxf(xv.z + ev.z, 0.0f);
  m.w = fmaxf(xv.w + ev.w, 0.0f);

  float* hp = h + (long long)dst * DIM + lane * 4;
  atomicAdd(hp + 0, m.x);
  atomicAdd(hp + 1, m.y);
  atomicAdd(hp + 2, m.z);
  atomicAdd(hp + 3, m.w);
}

// ---------------------------------------------------------------------------
// Kernels 3/4: 128x128 fp32 GEMM via V_WMMA_F32_16X16X4_F32.
//   out[rows, 0:128] = op( A[rows, 0:128] @ W + bias )
//   DO_LN=true : op = ReLU(LayerNorm(.)) with gamma/beta  (MLP layer 1)
//   DO_LN=false: op = identity                             (MLP layer 2)
// Block = 256 threads = 8 waves; each wave owns a 16-row tile (128 rows/block).
// W is staged transposed in LDS (row stride 130 floats to spread banks).
//
// A-fragment layout (16x4 f32): lanes 0-15 carry K = k0,k0+1 of row (lane&15),
// lanes 16-31 carry K = k0+2,k0+3  -> per-lane base offset 2*(lane>>4).
// C/D layout: element (M = j + 8*(lane>>4), Ncol = tile*16 + (lane&15)) in acc[j].
// ---------------------------------------------------------------------------
template <bool DO_LN>
__global__ void k_mlp(const float* __restrict__ A,
                      const float* __restrict__ W,
                      const float* __restrict__ bias,
                      const float* __restrict__ ln_g,
                      const float* __restrict__ ln_b,
                      float* __restrict__ out) {
  extern __shared__ float sWt[];            // [128][130] transposed W
  const int tid = threadIdx.x;

  // Stage W transposed: sWt[col][k] = W[k][col]
  for (int i = tid; i < DIM * DIM; i += 256) {
    int k = i >> 7;
    int c = i & 127;
    sWt[c * (DIM + 2) + k] = W[i];
  }
  __syncthreads();

  const int lane = tid & 31;
  const int wave = tid >> 5;
  const int half = lane >> 4;   // 0: K%4 in {0,1}, 1: K%4 in {2,3}
  const int l16  = lane & 15;
  const long long rowBase = (long long)blockIdx.x * 128 + wave * 16;

  // Load this lane's 64 A values (its row, its K-parity) into registers first,
  // so in-place A==out is safe (wave owns its 16 rows exclusively).
  long long arow = rowBase + l16;
  if (arow >= N_NODES) arow = N_NODES - 1;      // clamp; clamped rows never stored
  const float* ap = A + arow * DIM + 2 * half;
  v2f af[32];
#pragma unroll
  for (int i = 0; i < 32; ++i) {
    float2 t = *(const float2*)(ap + i * 4);
    af[i].x = t.x;
    af[i].y = t.y;
  }

  v8f acc[8];
#pragma unroll
  for (int t = 0; t < 8; ++t) {
    v8f c = {0.f, 0.f, 0.f, 0.f, 0.f, 0.f, 0.f, 0.f};
    const float* bp = sWt + (t * 16 + l16) * (DIM + 2) + 2 * half;
#pragma unroll
    for (int k = 0; k < 32; ++k) {
      float2 bb = *(const float2*)(bp + k * 4);   // ds_load_b64
      v2f b;
      b.x = bb.x;
      b.y = bb.y;
      // D = A(16x4) * B(4x16) + C   (fp32 WMMA, K advances by 4 each call)
      c = __builtin_amdgcn_wmma_f32_16x16x4_f32(
              /*neg_a=*/false, af[k], /*neg_b=*/false, b,
              /*c_mod=*/(short)0, c, /*reuse_a=*/false, /*reuse_b=*/false);
    }
    float bv = bias[t * 16 + l16];                // per-column bias
#pragma unroll
    for (int j = 0; j < 8; ++j) c[j] += bv;
    acc[t] = c;
  }

  if (DO_LN) {
    // Row statistics: row M = j + 8*half lives in acc[*][j] across lanes of its
    // 16-lane half. xor-shuffles with masks 1,2,4,8 reduce within each half.
#pragma unroll
    for (int j = 0; j < 8; ++j) {
      float s = 0.0f, q = 0.0f;
#pragma unroll
      for (int t = 0; t < 8; ++t) {
        float v = acc[t][j];
        s += v;
        q += v * v;
      }
#pragma unroll
      for (int off = 1; off <= 8; off <<= 1) {
        s += __shfl_xor(s, off, 32);
        q += __shfl_xor(q, off, 32);
      }
      float mean = s * (1.0f / DIM);
      float var  = q * (1.0f / DIM) - mean * mean;
      float rstd = rsqrtf(var + LN_EPS);
#pragma unroll
      for (int t = 0; t < 8; ++t)
        acc[t][j] = (acc[t][j] - mean) * rstd;
    }
    // gamma/beta (per column) + ReLU
#pragma unroll
    for (int t = 0; t < 8; ++t) {
      int col = t * 16 + l16;
      float gg = ln_g[col];
      float bb = ln_b[col];
#pragma unroll
      for (int j = 0; j < 8; ++j)
        acc[t][j] = fmaxf(acc[t][j] * gg + bb, 0.0f);
    }
  }

  // Store: element (row = rowBase + j + 8*half, col = t*16 + l16)
#pragma unroll
  for (int j = 0; j < 8; ++j) {
    long long row = rowBase + j + 8 * half;
    if (row < N_NODES) {
#pragma unroll
      for (int t = 0; t < 8; ++t)
        out[row * DIM + t * 16 + l16] = acc[t][j];
    }
  }
}

// ---------------------------------------------------------------------------
extern "C" void kernel_launch(void* const* d_in, const int* in_sizes, int n_in,
                              void* d_out, int out_size, void* d_ws, size_t ws_size,
                              hipStream_t stream) {
  const float* x   = (const float*)d_in[0];
  const int*   ei  = (const int*)  d_in[1];
  const int*   ea  = (const int*)  d_in[2];
  const float* emb = (const float*)d_in[3];
  const float* eps = (const float*)d_in[4];
  const float* W1  = (const float*)d_in[5];
  const float* b1  = (const float*)d_in[6];
  const float* lg  = (const float*)d_in[7];
  const float* lb  = (const float*)d_in[8];
  const float* W2  = (const float*)d_in[9];
  const float* b2  = (const float*)d_in[10];
  float* out = (float*)d_out;
  float* h   = (float*)d_ws;     // N_NODES*128 f32 scratch (h, then h1 in-place)

  (void)in_sizes; (void)n_in; (void)out_size; (void)ws_size;

  // 1) h = (1+eps)*x
  {
    long long n4 = (long long)N_NODES * DIM / 4;
    int blocks = (int)((n4 + 255) / 256);
    k_init<<<blocks, 256, 0, stream>>>(x, eps, h);
  }
  // 2) edge message + scatter-add (one wave per edge)
  {
    long long threads = (long long)N_EDGES * 32;
    int blocks = (int)((threads + 255) / 256);
    k_scatter<<<blocks, 256, 0, stream>>>(x, ei, ea, emb, h);
  }
  // 3) h1 = ReLU(LN(h @ W1 + b1))  — in place on h
  // 4) out = h1 @ W2 + b2
  {
    int gb = (N_NODES + 127) / 128;
    size_t shmem = (size_t)DIM * (DIM + 2) * sizeof(float);   // 66,560 B < 320 KB
    k_mlp<true ><<<gb, 256, shmem, stream>>>(h, W1, b1, lg, lb, h);
    k_mlp<false><<<gb, 256, shmem, stream>>>(h, W2, b2, nullptr, nullptr, out);
  }
}